// DIST_plus_46677704573289
// MI455X (gfx1250) — compile-verified
//
#include <hip/hip_runtime.h>
#include <hip/hip_bf16.h>
#include <math.h>

#define BATCH 4096
#define CLS   16000
#define NTOP  10
#define EPS_F 1e-8f

typedef float v2f __attribute__((ext_vector_type(2)));
typedef float v8f __attribute__((ext_vector_type(8)));

// ---------------------------------------------------------------------------
// Kernel A: per-row softmax stats (max, 1/sumexp, first-argmax) for both
// matrices (blockIdx.y selects z_s / z_t). One block (256 thr = 8 waves) / row.
// ---------------------------------------------------------------------------
__global__ void row_stats_kernel(const float* __restrict__ zs,
                                 const float* __restrict__ zt,
                                 float* __restrict__ rowmax,   // [2*BATCH]
                                 float* __restrict__ rowinv,   // [2*BATCH]
                                 int*   __restrict__ jarg) {   // [2*BATCH]
  const int b = blockIdx.x;
  const int which = blockIdx.y;
  const float* row = (which == 0 ? zs : zt) + (size_t)b * CLS;

  __shared__ float sv[256];
  __shared__ int   si[256];
  const int t = threadIdx.x;

  float m = -INFINITY; int mi = 0x7fffffff;
  for (int k = t; k < CLS; k += 256) {
    float v = row[k];
    if (v > m) { m = v; mi = k; }
    else if (v == m && k < mi) { mi = k; }
  }
  sv[t] = m; si[t] = mi;
  __syncthreads();
  for (int s = 128; s > 0; s >>= 1) {
    if (t < s) {
      float ov = sv[t + s]; int oi = si[t + s];
      if (ov > sv[t] || (ov == sv[t] && oi < si[t])) { sv[t] = ov; si[t] = oi; }
    }
    __syncthreads();
  }
  const float rmax = sv[0];
  const int   rj   = si[0];
  __syncthreads();

  float acc = 0.0f;
  for (int k = t; k < CLS; k += 256) acc += __expf(row[k] - rmax);
  sv[t] = acc;
  __syncthreads();
  for (int s = 128; s > 0; s >>= 1) {
    if (t < s) sv[t] += sv[t + s];
    __syncthreads();
  }
  if (t == 0) {
    rowmax[which * BATCH + b] = rmax;
    rowinv[which * BATCH + b] = 1.0f / sv[0];
    jarg  [which * BATCH + b] = rj;
  }
}

// ---------------------------------------------------------------------------
// Kernel B: per-row stable-sort positions of columns 0..9 (rank-clamp trick).
// s_c = #{k : z_k < z_c} + #{k < c : z_k == z_c}.  One block per (row, matrix).
// ---------------------------------------------------------------------------
__global__ void rank_count_kernel(const float* __restrict__ zs,
                                  const float* __restrict__ zt,
                                  int* __restrict__ cnt) {     // [2][BATCH][NTOP]
  const int b = blockIdx.x;
  const int which = blockIdx.y;
  const float* row = (which == 0 ? zs : zt) + (size_t)b * CLS;

  __shared__ float piv[NTOP];
  __shared__ int   scnt[NTOP];
  const int t = threadIdx.x;
  if (t < NTOP) { piv[t] = row[t]; scnt[t] = 0; }
  __syncthreads();

  float pv[NTOP];
#pragma unroll
  for (int c = 0; c < NTOP; ++c) pv[c] = piv[c];

  int lc[NTOP];
#pragma unroll
  for (int c = 0; c < NTOP; ++c) lc[c] = 0;

  for (int k = t; k < CLS; k += 256) {
    __builtin_prefetch(row + k + 2048, 0, 1);
    const float v = row[k];
#pragma unroll
    for (int c = 0; c < NTOP; ++c) {
      lc[c] += (v < pv[c]) ? 1 : ((v == pv[c] && k < c) ? 1 : 0);
    }
  }
#pragma unroll
  for (int c = 0; c < NTOP; ++c) atomicAdd(&scnt[c], lc[c]);
  __syncthreads();
  if (t < NTOP) cnt[((size_t)which * BATCH + b) * NTOP + t] = scnt[t];
}

// ---------------------------------------------------------------------------
// Kernel C: intra-class column statistics via V_WMMA_F32_16X16X4_F32.
// A = all-ones 16x4  =>  D[m][n] = sum_k B[k][n]  (4-row column sums / op).
// Each wave owns 16 columns; 5 independent accumulators keep the XDL pipe full.
// Finishes by writing per-column pearson.
// ---------------------------------------------------------------------------
__global__ void intra_wmma_kernel(const float* __restrict__ zs,
                                  const float* __restrict__ zt,
                                  const float* __restrict__ rowmax,  // [2*BATCH]
                                  const float* __restrict__ rowinv,  // [2*BATCH]
                                  float* __restrict__ pearson_col) { // [CLS]
  const int lane = threadIdx.x & 31;
  const int wave = threadIdx.x >> 5;
  const int col  = blockIdx.x * 128 + wave * 16 + (lane & 15);
  const int rsel = (lane >> 4) * 2;          // lanes 0-15 -> rows {0,1}; 16-31 -> {2,3}

  const float* rmax_s = rowmax;
  const float* rmax_t = rowmax + BATCH;
  const float* rinv_s = rowinv;
  const float* rinv_t = rowinv + BATCH;

  const v2f ones = {1.0f, 1.0f};
  v8f acc_s  = {};
  v8f acc_t  = {};
  v8f acc_st = {};
  v8f acc_ss = {};
  v8f acc_tt = {};

  for (int chunk = 0; chunk < BATCH; chunk += 4) {
    const int r0 = chunk + rsel;
    const int r1 = r0 + 1;
    const size_t i0 = (size_t)r0 * CLS + col;
    const size_t i1 = (size_t)r1 * CLS + col;
    __builtin_prefetch(zs + i0 + 4 * (size_t)CLS, 0, 1);
    __builtin_prefetch(zt + i0 + 4 * (size_t)CLS, 0, 1);

    const float ys0 = __expf(zs[i0] - rmax_s[r0]) * rinv_s[r0];
    const float ys1 = __expf(zs[i1] - rmax_s[r1]) * rinv_s[r1];
    const float yt0 = __expf(zt[i0] - rmax_t[r0]) * rinv_t[r0];
    const float yt1 = __expf(zt[i1] - rmax_t[r1]) * rinv_t[r1];

    const v2f bs  = {ys0, ys1};
    const v2f bt  = {yt0, yt1};
    const v2f bst = {ys0 * yt0, ys1 * yt1};
    const v2f bss = {ys0 * ys0, ys1 * ys1};
    const v2f btt = {yt0 * yt0, yt1 * yt1};

    acc_s  = __builtin_amdgcn_wmma_f32_16x16x4_f32(false, ones, false, bs,  (short)0, acc_s,  false, false);
    acc_t  = __builtin_amdgcn_wmma_f32_16x16x4_f32(false, ones, false, bt,  (short)0, acc_t,  false, false);
    acc_st = __builtin_amdgcn_wmma_f32_16x16x4_f32(false, ones, false, bst, (short)0, acc_st, false, false);
    acc_ss = __builtin_amdgcn_wmma_f32_16x16x4_f32(false, ones, false, bss, (short)0, acc_ss, false, false);
    acc_tt = __builtin_amdgcn_wmma_f32_16x16x4_f32(false, ones, false, btt, (short)0, acc_tt, false, false);
  }

  // Every row of D holds the full column sum; lanes 0-15 cover all 16 columns.
  if ((lane >> 4) == 0) {
    const float Bf = (float)BATCH;
    const float sum_s  = acc_s[0];
    const float sum_t  = acc_t[0];
    const float sum_st = acc_st[0];
    const float sum_ss = acc_ss[0];
    const float sum_tt = acc_tt[0];
    const float ms = sum_s / Bf;
    const float mt = sum_t / Bf;
    const float num = sum_st - Bf * ms * mt;
    const float na  = sqrtf(fmaxf(sum_ss - Bf * ms * ms, 0.0f));
    const float nb  = sqrtf(fmaxf(sum_tt - Bf * mt * mt, 0.0f));
    pearson_col[col] = num / (na * nb + EPS_F);
  }
}

// ---------------------------------------------------------------------------
// Kernel D: closed-form per-row spearman from the sparse rank representation.
// fs = 10 everywhere except position p_c (value c), entry dropped iff s_c == j.
// ---------------------------------------------------------------------------
__global__ void inter_row_kernel(const int* __restrict__ cnt,   // [2][BATCH][NTOP]
                                 const int* __restrict__ jarg,  // [2*BATCH]
                                 float* __restrict__ p_row,
                                 float* __restrict__ eq_row) {
  const int b = blockIdx.x * blockDim.x + threadIdx.x;
  if (b >= BATCH) return;

  const int js = jarg[b];
  const int jt = jarg[BATCH + b];

  int ps[NTOP], pt[NTOP];
  bool okS[NTOP], okT[NTOP];
  float su = 0.f, su2 = 0.f, sv = 0.f, sv2 = 0.f;
#pragma unroll
  for (int c = 0; c < NTOP; ++c) {
    const int s = cnt[(size_t)b * NTOP + c];
    okS[c] = (s != js);
    ps[c]  = s - ((s > js) ? 1 : 0);
    if (okS[c]) { const float u = (float)c - 10.0f; su += u; su2 += u * u; }
    const int tt = cnt[((size_t)BATCH + b) * NTOP + c];
    okT[c] = (tt != jt);
    pt[c]  = tt - ((tt > jt) ? 1 : 0);
    if (okT[c]) { const float v = (float)c - 10.0f; sv += v; sv2 += v * v; }
  }
  float cross = 0.f;
#pragma unroll
  for (int c = 0; c < NTOP; ++c) {
    if (!okS[c]) continue;
#pragma unroll
    for (int d = 0; d < NTOP; ++d) {
      if (okT[d] && ps[c] == pt[d]) cross += ((float)c - 10.0f) * ((float)d - 10.0f);
    }
  }
  const float N = (float)(CLS - 1);
  const float num = cross - su * sv / N;
  const float na  = sqrtf(fmaxf(su2 - su * su / N, 0.0f));
  const float nb  = sqrtf(fmaxf(sv2 - sv * sv / N, 0.0f));
  p_row[b]  = num / (na * nb + EPS_F);
  eq_row[b] = (js == jt) ? 1.0f : 0.0f;
}

// ---------------------------------------------------------------------------
// Kernel E: deterministic final reduction -> scalar loss.
// ---------------------------------------------------------------------------
__global__ void final_reduce_kernel(const float* __restrict__ pcol,
                                    const float* __restrict__ p_row,
                                    const float* __restrict__ eq_row,
                                    float* __restrict__ out) {
  __shared__ float s0[256], s1[256], s2[256];
  const int t = threadIdx.x;
  float a = 0.f, b = 0.f, c = 0.f;
  for (int k = t; k < CLS; k += 256) a += pcol[k];
  for (int k = t; k < BATCH; k += 256) { b += p_row[k]; c += eq_row[k]; }
  s0[t] = a; s1[t] = b; s2[t] = c;
  __syncthreads();
  for (int s = 128; s > 0; s >>= 1) {
    if (t < s) { s0[t] += s0[t + s]; s1[t] += s1[t + s]; s2[t] += s2[t + s]; }
    __syncthreads();
  }
  if (t == 0) {
    const float intra = 1.0f - s0[0] / (float)CLS;
    const float inter = 1.0f - (s2[0] / (float)BATCH + s1[0] / (float)BATCH);
    out[0] = inter + intra;   // BETA = GAMMA = TAU = 1
  }
}

// ---------------------------------------------------------------------------
extern "C" void kernel_launch(void* const* d_in, const int* in_sizes, int n_in,
                              void* d_out, int out_size, void* d_ws, size_t ws_size,
                              hipStream_t stream) {
  const float* zs = (const float*)d_in[0];
  const float* zt = (const float*)d_in[1];
  float* out = (float*)d_out;

  float* ws      = (float*)d_ws;
  float* rowmax  = ws;                       // 2*BATCH
  float* rowinv  = ws + 2 * BATCH;           // 2*BATCH
  int*   jarg    = (int*)(ws + 4 * BATCH);   // 2*BATCH
  int*   cnt     = (int*)(ws + 6 * BATCH);   // 2*BATCH*NTOP
  float* p_row   = ws + 26 * BATCH;          // BATCH
  float* eq_row  = ws + 27 * BATCH;          // BATCH
  float* pcol    = ws + 28 * BATCH;          // CLS

  row_stats_kernel<<<dim3(BATCH, 2), 256, 0, stream>>>(zs, zt, rowmax, rowinv, jarg);
  rank_count_kernel<<<dim3(BATCH, 2), 256, 0, stream>>>(zs, zt, cnt);
  intra_wmma_kernel<<<CLS / 128, 256, 0, stream>>>(zs, zt, rowmax, rowinv, pcol);
  inter_row_kernel<<<(BATCH + 255) / 256, 256, 0, stream>>>(cnt, jarg, p_row, eq_row);
  final_reduce_kernel<<<1, 256, 0, stream>>>(pcol, p_row, eq_row, out);
}